// MAMGCN_encblock_24953759990165
// MI455X (gfx1250) — compile-verified
//
#include <hip/hip_runtime.h>
#include <math.h>

// ---------------------------------------------------------------------------
// MAMGCN encoder block for MI455X (gfx1250, wave32, WMMA).
// All dense GEMMs use V_WMMA_F32_16X16X4_F32 via the clang builtin, with a
// branch-free, software-pipelined K loop (K must be a multiple of 4 — all
// call sites are: K in {4, 64, 128, 256, 300}).
//
// Input flattening assumption (setup_inputs() dict, recursive insertion order):
//  0:x 1:STE 2:adj
//  3..12  : mamba.fwd {in_W, conv_W, conv_b, xproj_W, dt_W, dt_b, A_log, D, out_W, rms_g}
// 13..22  : mamba.bwd (same 10)
// 23..26  : mamba {proj_W, proj_b, ln_g, ln_b}
// 27..38  : tat {qkv_W,qkv_b,o_W,o_b,f1_W,f1_b,f2_W,f2_b,ln1_g,ln1_b,ln2_g,ln2_b}
// 39..50  : gat (same 12)
// 51..58  : gcn {W0,b0,Wg0,bg0,W1,b1,Wg1,bg1}
// 59..64  : sffn {W1,b1,W2,b2,bn_g,bn_b}
// 65..70  : tffn (same 6)
// 71,72   : gate_W, gate_b
// ---------------------------------------------------------------------------

typedef __attribute__((ext_vector_type(2))) float v2f;
typedef __attribute__((ext_vector_type(8))) float v8f;

constexpr int Bc = 4, Nc = 300, Tc = 12, Hc = 8;
constexpr int ROWS = Bc * Nc * Tc;   // 14400
constexpr int SEQ  = Bc * Nc;        // 1200

constexpr size_t S64  = (size_t)ROWS * 64;
constexpr size_t S128 = (size_t)ROWS * 128;
constexpr size_t S192 = (size_t)ROWS * 192;
constexpr size_t S256 = (size_t)ROWS * 256;
constexpr size_t S36  = (size_t)ROWS * 36;

// persistent buffers (float offsets into d_ws)
constexpr size_t O_XENC = 0;
constexpr size_t O_SP   = O_XENC + S64;
constexpr size_t O_XM   = O_SP   + S64;
constexpr size_t O_XT   = O_XM   + S64;
constexpr size_t O_XG   = O_XT   + S64;   // gcn out (sp layout)
constexpr size_t O_XSAT = O_XG   + S64;   // gat out (sp layout)
constexpr size_t O_XS   = O_XSAT + S64;
constexpr size_t O_XTF  = O_XS   + S64;
constexpr size_t O_AHAT = O_XTF  + S64;   // 300*300
constexpr size_t O_STAT = O_AHAT + 90112; // 128 floats (mean|var)
constexpr size_t O_AR   = O_STAT + 128;   // phase-shared arena

// arena: mamba phase
constexpr size_t A_XR   = O_AR;
constexpr size_t A_XZ   = A_XR  + S64;
constexpr size_t A_XB   = A_XZ  + S256;
constexpr size_t A_DBL  = A_XB  + S128;
constexpr size_t A_DT   = A_DBL + S36;
constexpr size_t A_YSC  = A_DT  + S128;
constexpr size_t A_YF   = A_YSC + S128;
constexpr size_t A_YB   = A_YF  + S64;
constexpr size_t A_CAT  = A_YB  + S64;
constexpr size_t A_YPROJ= A_CAT + S128;
// arena: transformer phase (reuses the same region, sequential phases)
constexpr size_t A_QKV  = O_AR;
constexpr size_t A_ATTO = A_QKV + S192;
constexpr size_t A_OPROJ= A_ATTO+ S64;
constexpr size_t A_XMID = A_OPROJ+S64;
constexpr size_t A_FFH  = A_XMID+ S64;
constexpr size_t A_F2O  = A_FFH + S256;
// arena: gcn phase
constexpr size_t A_T0   = O_AR;
constexpr size_t A_M0   = A_T0 + S64;
constexpr size_t A_G0   = A_M0 + S64;
constexpr size_t A_H1   = A_G0 + S64;
// arena: tfn / gate phase
constexpr size_t A_CAT2 = O_AR;
constexpr size_t A_O1   = A_CAT2 + S128;
constexpr size_t A_O2   = A_O1   + S128;
constexpr size_t A_CATG = O_AR;

// ------------------------------- device helpers ----------------------------

__device__ __forceinline__ float act_f(float v, int act) {
  switch (act) {
    case 1: return fmaxf(v, 0.f);                      // relu
    case 2: return v / (1.f + __expf(-v));             // silu
    case 3: return (v > 20.f) ? v : log1pf(__expf(v)); // softplus
    case 4: return 1.f / (1.f + __expf(-v));           // sigmoid
  }
  return v;
}

// ------------------------------- WMMA GEMM ---------------------------------
// C[batch] = act(A[batch] (MxK) @ B[batch] (KxN) + bias) (+res)
// One wave per 16x16 output tile; fp32 WMMA 16x16x4; branch-free pipelined
// K loop. Requires K % 4 == 0.
//
// Edge tiles: out-of-range lanes clamp their load address to the last valid
// row/col. Since A-fragment row m feeds only output row m and B-fragment
// col n feeds only output col n, the clamped (duplicate) values land only in
// output elements the epilogue never stores. EXEC stays all-1s for WMMA.
__global__ void gemm_wmma_k(const float* __restrict__ A, const float* __restrict__ B,
                            const float* __restrict__ bias, const float* __restrict__ res,
                            float* __restrict__ C, int M, int N, int K,
                            int lda, int ldb, int ldc,
                            long long sA, long long sB, long long sC, int act) {
  const int batch = blockIdx.z;
  A += (size_t)batch * sA;
  B += (size_t)batch * sB;
  C += (size_t)batch * sC;
  if (res) res += (size_t)batch * sC;

  const int lane = threadIdx.x;           // 0..31 (wave32)
  const int ml   = lane & 15;
  const int khi  = (lane >> 4) << 1;      // 0 or 2
  const int row  = blockIdx.y * 16 + ml;  // A fragment row
  const int col  = blockIdx.x * 16 + ml;  // B fragment col

  // clamped fragment base pointers (branch-free OOB handling)
  const float* Ap = A + (size_t)min(row, M - 1) * lda + khi;
  const float* Bp = B + min(col, N - 1) + (size_t)khi * ldb;

  v8f acc = {0.f, 0.f, 0.f, 0.f, 0.f, 0.f, 0.f, 0.f};

  // prologue fragment (lda, khi even everywhere -> 8B-aligned v2f load)
  v2f a = *(const v2f*)Ap;
  v2f b;
  b.x = Bp[0];
  b.y = Bp[ldb];

#pragma unroll 2
  for (int k0 = 4; k0 < K; k0 += 4) {
    // prefetch next fragments before consuming current ones (SW pipeline)
    v2f an = *(const v2f*)(Ap + k0);
    v2f bn;
    bn.x = Bp[(size_t)k0 * ldb];
    bn.y = Bp[(size_t)(k0 + 1) * ldb];
    acc = __builtin_amdgcn_wmma_f32_16x16x4_f32(false, a, false, b,
                                                (short)0, acc, false, false);
    a = an; b = bn;
  }
  acc = __builtin_amdgcn_wmma_f32_16x16x4_f32(false, a, false, b,
                                              (short)0, acc, false, false);

  const int oc = blockIdx.x * 16 + ml;
  if (oc >= N) return;
  const float bv = bias ? bias[oc] : 0.f;
  const int rbase = blockIdx.y * 16 + ((lane >> 4) << 3); // lanes 16-31 -> M+8
#pragma unroll
  for (int i = 0; i < 8; i++) {
    const int orow = rbase + i;
    if (orow < M) {
      float v = act_f(acc[i] + bv, act);
      if (res) v += res[(size_t)orow * ldc + oc];
      C[(size_t)orow * ldc + oc] = v;
    }
  }
}

// ------------------------------ small kernels ------------------------------

__global__ void add_k(const float* a, const float* b, float* o, int n) {
  int i = blockIdx.x * blockDim.x + threadIdx.x;
  if (i < n) o[i] = a[i] + b[i];
}

// (B,N,T,C) -> (B,T,N,C)
__global__ void tr_to_sp_k(const float* __restrict__ x, float* __restrict__ sp, int n) {
  int i = blockIdx.x * blockDim.x + threadIdx.x;
  if (i >= n) return;
  int c = i & 63, r = i >> 6;
  int t = r % Tc, nn = (r / Tc) % Nc, b = r / (Tc * Nc);
  sp[(((size_t)(b * Tc + t) * Nc + nn) << 6) + c] = x[i];
}

// RMSNorm rows of 64; rev flips time within each sequence of length T
__global__ void rms_k(const float* __restrict__ x, const float* __restrict__ g,
                      float* __restrict__ o, int T, int rev) {
  int r = blockIdx.x;
  int t = r % T, s = r / T;
  int src = rev ? (s * T + (T - 1 - t)) : r;
  int c = threadIdx.x;
  float v = x[(size_t)src * 64 + c];
  __shared__ float sm[64];
  sm[c] = v * v; __syncthreads();
  for (int st = 32; st > 0; st >>= 1) { if (c < st) sm[c] += sm[c + st]; __syncthreads(); }
  float mean = sm[0] * (1.f / 64.f);
  o[(size_t)r * 64 + c] = v * rsqrtf(mean + 1e-5f) * g[c];
}

// LayerNorm of (a + b) over 64 channels
__global__ void ln_k(const float* __restrict__ a, const float* __restrict__ b,
                     const float* __restrict__ g, const float* __restrict__ be,
                     float* __restrict__ o) {
  int r = blockIdx.x, c = threadIdx.x;
  float v = a[(size_t)r * 64 + c] + (b ? b[(size_t)r * 64 + c] : 0.f);
  __shared__ float s1[64], s2[64];
  s1[c] = v; s2[c] = v * v; __syncthreads();
  for (int st = 32; st > 0; st >>= 1) {
    if (c < st) { s1[c] += s1[c + st]; s2[c] += s2[c + st]; }
    __syncthreads();
  }
  float m = s1[0] * (1.f / 64.f);
  float var = s2[0] * (1.f / 64.f) - m * m;
  o[(size_t)r * 64 + c] = (v - m) * rsqrtf(var + 1e-5f) * g[c] + be[c];
}

// depthwise causal conv (DC=4) over time + silu; xz holds [x|z] (256 wide)
__global__ void conv_silu_k(const float* __restrict__ xz, const float* __restrict__ cw,
                            const float* __restrict__ cb, float* __restrict__ xb,
                            int total, int T) {
  int i = blockIdx.x * blockDim.x + threadIdx.x;
  if (i >= total) return;
  int d = i & 127, t = (i >> 7) % T, s = i / (128 * T);
  float v = cb[d];
#pragma unroll
  for (int k = 0; k < 4; k++) {
    int tt = t - 3 + k;
    if (tt >= 0) v += cw[d * 4 + k] * xz[((size_t)(s * T + tt)) * 256 + d];
  }
  xb[i] = v / (1.f + __expf(-v));
}

// selective scan: thread per (sequence, channel d); state h[16]
__global__ void scan_k(const float* __restrict__ dt, const float* __restrict__ xb,
                       const float* __restrict__ dbl, const float* __restrict__ xz,
                       const float* __restrict__ Alog, const float* __restrict__ Dp,
                       float* __restrict__ y, int nseq, int T) {
  int i = blockIdx.x * blockDim.x + threadIdx.x;
  if (i >= nseq * 128) return;
  int d = i & 127, s = i >> 7;
  float nA[16], h[16];
#pragma unroll
  for (int j = 0; j < 16; j++) { nA[j] = -__expf(Alog[d * 16 + j]); h[j] = 0.f; }
  const float Dv = Dp[d];
  for (int t = 0; t < T; t++) {
    size_t r = (size_t)s * T + t;
    float dtv = dt[r * 128 + d];
    float xbv = xb[r * 128 + d];
    float accv = 0.f;
#pragma unroll
    for (int j = 0; j < 16; j++) {
      float dA = __expf(dtv * nA[j]);
      float bx = dtv * dbl[r * 36 + 4 + j] * xbv;
      h[j] = dA * h[j] + bx;
      accv += h[j] * dbl[r * 36 + 20 + j];
    }
    float yv = accv + xbv * Dv;
    float zv = xz[r * 256 + 128 + d];
    y[r * 128 + d] = yv * (zv / (1.f + __expf(-zv)));
  }
}

// cat[:, :64] = x + yf ; cat[:, 64:] = x + time-reversed(yb)
__global__ void mamba_cat_k(const float* __restrict__ x, const float* __restrict__ yf,
                            const float* __restrict__ yb, float* __restrict__ cat,
                            int total, int T) {
  int i = blockIdx.x * blockDim.x + threadIdx.x;
  if (i >= total) return;
  int c = i & 127, r = i >> 7;
  int t = r % T, s = r / T;
  if (c < 64) cat[i] = x[(size_t)r * 64 + c] + yf[(size_t)r * 64 + c];
  else {
    int rb = s * T + (T - 1 - t);
    cat[i] = x[(size_t)r * 64 + (c - 64)] + yb[(size_t)rb * 64 + (c - 64)];
  }
}

// fused attention: one wave per (batch, head, query); dh = 8
__global__ void attn_k(const float* __restrict__ qkv, const float* __restrict__ adj,
                       float* __restrict__ o, int L, int useMask) {
  const int q = blockIdx.x, h = blockIdx.y, b = blockIdx.z;
  const int lane = threadIdx.x;
  const float scale = 0.35355339059327373f; // 1/sqrt(8)
  const float* base = qkv + (size_t)b * L * 192;
  float qv[8];
#pragma unroll
  for (int d = 0; d < 8; d++) qv[d] = base[(size_t)q * 192 + h * 8 + d];
  float sv[10];
  float mx = -3.0e38f;
  int cnt = 0;
  for (int k = lane; k < L; k += 32) {
    const float* kp = base + (size_t)k * 192 + 64 + h * 8;
    float s = 0.f;
#pragma unroll
    for (int d = 0; d < 8; d++) s += qv[d] * kp[d];
    s *= scale;
    if (useMask && !(adj[(size_t)q * L + k] > 0.f)) s = -1e9f;
    sv[cnt++] = s;
    mx = fmaxf(mx, s);
  }
#pragma unroll
  for (int m = 16; m >= 1; m >>= 1) mx = fmaxf(mx, __shfl_xor(mx, m, 32));
  float ps = 0.f, ov[8];
#pragma unroll
  for (int d = 0; d < 8; d++) ov[d] = 0.f;
  int idx = 0;
  for (int k = lane; k < L; k += 32) {
    float p = __expf(sv[idx++] - mx);
    ps += p;
    const float* vp = base + (size_t)k * 192 + 128 + h * 8;
#pragma unroll
    for (int d = 0; d < 8; d++) ov[d] += p * vp[d];
  }
#pragma unroll
  for (int m = 16; m >= 1; m >>= 1) ps += __shfl_xor(ps, m, 32);
#pragma unroll
  for (int d = 0; d < 8; d++) {
    float t = ov[d];
#pragma unroll
    for (int m = 16; m >= 1; m >>= 1) t += __shfl_xor(t, m, 32);
    ov[d] = t;
  }
  if (lane == 0) {
    float inv = 1.f / ps;
#pragma unroll
    for (int d = 0; d < 8; d++) o[((size_t)b * L + q) * 64 + h * 8 + d] = ov[d] * inv;
  }
}

// A_hat = adj / max(rowsum, 1e-6)
__global__ void ahat_k(const float* __restrict__ adj, float* __restrict__ ah, int Nn) {
  int r = blockIdx.x, tid = threadIdx.x;
  __shared__ float sm[256];
  float s = 0.f;
  for (int j = tid; j < Nn; j += 256) s += adj[(size_t)r * Nn + j];
  sm[tid] = s; __syncthreads();
  for (int st = 128; st > 0; st >>= 1) { if (tid < st) sm[tid] += sm[tid + st]; __syncthreads(); }
  float inv = 1.f / fmaxf(sm[0], 1e-6f);
  for (int j = tid; j < Nn; j += 256) ah[(size_t)r * Nn + j] = adj[(size_t)r * Nn + j] * inv;
}

// h' = g*m + (1-g)*h   (m is already relu(A_hat@..+b))
__global__ void gcomb_k(const float* g, const float* m, const float* h, float* o, int n) {
  int i = blockIdx.x * blockDim.x + threadIdx.x;
  if (i < n) { float gv = g[i]; o[i] = gv * m[i] + (1.f - gv) * h[i]; }
}

// per-channel mean/var over all rows (ddof=0)
__global__ void chstat_k(const float* __restrict__ x, float* __restrict__ st, int rows) {
  int c = blockIdx.x, tid = threadIdx.x;
  __shared__ float s1[256], s2[256];
  float a = 0.f, b = 0.f;
  for (int r = tid; r < rows; r += 256) {
    float v = x[(size_t)r * 64 + c];
    a += v; b += v * v;
  }
  s1[tid] = a; s2[tid] = b; __syncthreads();
  for (int stp = 128; stp > 0; stp >>= 1) {
    if (tid < stp) { s1[tid] += s1[tid + stp]; s2[tid] += s2[tid + stp]; }
    __syncthreads();
  }
  if (tid == 0) {
    float m = s1[0] / rows;
    st[c] = m;
    st[64 + c] = s2[0] / rows - m * m;
  }
}

__global__ void bn_apply_k(const float* __restrict__ x, const float* __restrict__ st,
                           const float* __restrict__ g, const float* __restrict__ be,
                           float* __restrict__ o, int n) {
  int i = blockIdx.x * blockDim.x + threadIdx.x;
  if (i >= n) return;
  int c = i & 63;
  o[i] = (x[i] - st[c]) * rsqrtf(st[64 + c] + 1e-5f) * g[c] + be[c];
}

// cat of two row-layout (rows x 64) buffers into rows x 128
__global__ void cat_rows_k(const float* a, const float* b, float* cat, int total) {
  int i = blockIdx.x * blockDim.x + threadIdx.x;
  if (i >= total) return;
  int c = i & 127, r = i >> 7;
  cat[i] = (c < 64) ? a[(size_t)r * 64 + c] : b[(size_t)r * 64 + (c - 64)];
}

// cat of two sp-layout buffers into (B,N,T)-row-layout rows x 128
__global__ void cat_sp_k(const float* a, const float* b, float* cat, int total) {
  int i = blockIdx.x * blockDim.x + threadIdx.x;
  if (i >= total) return;
  int c = i & 127, r = i >> 7;
  int t = r % Tc, nn = (r / Tc) % Nc, bb = r / (Tc * Nc);
  size_t sp = ((size_t)(bb * Tc + t) * Nc + nn) * 64;
  cat[i] = (c < 64) ? a[sp + c] : b[sp + (c - 64)];
}

// ------------------------------- host helpers ------------------------------

static void launch_gemm(hipStream_t st, const float* A, const float* B, const float* bias,
                        const float* res, float* C, int M, int N, int K,
                        int lda, int ldb, int ldc, int batch,
                        long long sA, long long sB, long long sC, int act) {
  dim3 g((unsigned)((N + 15) / 16), (unsigned)((M + 15) / 16), (unsigned)batch);
  gemm_wmma_k<<<g, dim3(32), 0, st>>>(A, B, bias, res, C, M, N, K, lda, ldb, ldc,
                                      sA, sB, sC, act);
}

static inline int ceil_div(int a, int b) { return (a + b - 1) / b; }

// one Mamba direction: params at d_in[pb..pb+9]; writes (rows x 64) into ymb
static void run_mamba(hipStream_t st, const float* x, void* const* d_in, int pb, int rev,
                      float* ws, float* ymb) {
  const float* inW   = (const float*)d_in[pb + 0];
  const float* convW = (const float*)d_in[pb + 1];
  const float* convB = (const float*)d_in[pb + 2];
  const float* xprW  = (const float*)d_in[pb + 3];
  const float* dtW   = (const float*)d_in[pb + 4];
  const float* dtB   = (const float*)d_in[pb + 5];
  const float* Alog  = (const float*)d_in[pb + 6];
  const float* Dp    = (const float*)d_in[pb + 7];
  const float* outW  = (const float*)d_in[pb + 8];
  const float* rmsg  = (const float*)d_in[pb + 9];
  float *XR = ws + A_XR, *XZ = ws + A_XZ, *XB = ws + A_XB, *DBL = ws + A_DBL;
  float *DT = ws + A_DT, *YSC = ws + A_YSC;

  rms_k<<<ROWS, 64, 0, st>>>(x, rmsg, XR, Tc, rev);
  launch_gemm(st, XR, inW, nullptr, nullptr, XZ, ROWS, 256, 64, 64, 256, 256, 1, 0, 0, 0, 0);
  conv_silu_k<<<ceil_div(ROWS * 128, 256), 256, 0, st>>>(XZ, convW, convB, XB, ROWS * 128, Tc);
  launch_gemm(st, XB, xprW, nullptr, nullptr, DBL, ROWS, 36, 128, 128, 36, 36, 1, 0, 0, 0, 0);
  launch_gemm(st, DBL, dtW, dtB, nullptr, DT, ROWS, 128, 4, 36, 128, 128, 1, 0, 0, 0, 3);
  scan_k<<<ceil_div(SEQ * 128, 256), 256, 0, st>>>(DT, XB, DBL, XZ, Alog, Dp, YSC, SEQ, Tc);
  launch_gemm(st, YSC, outW, nullptr, nullptr, ymb, ROWS, 64, 128, 128, 64, 64, 1, 0, 0, 0, 0);
}

// transformer layer: params at d_in[pb..pb+11]; mask may be null
static void run_tf(hipStream_t st, const float* xin, int Bn, int L, void* const* d_in,
                   int pb, const float* mask, float* ws, float* out) {
  const float* qkvW = (const float*)d_in[pb + 0];
  const float* qkvB = (const float*)d_in[pb + 1];
  const float* oW   = (const float*)d_in[pb + 2];
  const float* oB   = (const float*)d_in[pb + 3];
  const float* f1W  = (const float*)d_in[pb + 4];
  const float* f1B  = (const float*)d_in[pb + 5];
  const float* f2W  = (const float*)d_in[pb + 6];
  const float* f2B  = (const float*)d_in[pb + 7];
  const float* l1g  = (const float*)d_in[pb + 8];
  const float* l1b  = (const float*)d_in[pb + 9];
  const float* l2g  = (const float*)d_in[pb + 10];
  const float* l2b  = (const float*)d_in[pb + 11];
  float *QKV = ws + A_QKV, *ATTO = ws + A_ATTO, *OPROJ = ws + A_OPROJ;
  float *XMID = ws + A_XMID, *FFH = ws + A_FFH, *F2O = ws + A_F2O;
  const int rows = Bn * L;

  launch_gemm(st, xin, qkvW, qkvB, nullptr, QKV, rows, 192, 64, 64, 192, 192, 1, 0, 0, 0, 0);
  attn_k<<<dim3((unsigned)L, Hc, (unsigned)Bn), 32, 0, st>>>(QKV, mask, ATTO, L, mask ? 1 : 0);
  launch_gemm(st, ATTO, oW, oB, nullptr, OPROJ, rows, 64, 64, 64, 64, 64, 1, 0, 0, 0, 0);
  ln_k<<<rows, 64, 0, st>>>(xin, OPROJ, l1g, l1b, XMID);
  launch_gemm(st, XMID, f1W, f1B, nullptr, FFH, rows, 256, 64, 64, 256, 256, 1, 0, 0, 0, 1);
  launch_gemm(st, FFH, f2W, f2B, nullptr, F2O, rows, 64, 256, 256, 64, 64, 1, 0, 0, 0, 0);
  ln_k<<<rows, 64, 0, st>>>(XMID, F2O, l2g, l2b, out);
}

// TFN fusion: cat -> relu GEMM -> GEMM -> global BN
static void run_tfn(hipStream_t st, const float* t1, const float* t2, int spLayout,
                    void* const* d_in, int pb, float* ws, float* out) {
  const float* W1 = (const float*)d_in[pb + 0];
  const float* b1 = (const float*)d_in[pb + 1];
  const float* W2 = (const float*)d_in[pb + 2];
  const float* b2 = (const float*)d_in[pb + 3];
  const float* bg = (const float*)d_in[pb + 4];
  const float* bb = (const float*)d_in[pb + 5];
  float *CAT2 = ws + A_CAT2, *O1 = ws + A_O1, *O2 = ws + A_O2, *STAT = ws + O_STAT;

  if (spLayout) cat_sp_k<<<ceil_div(ROWS * 128, 256), 256, 0, st>>>(t1, t2, CAT2, ROWS * 128);
  else          cat_rows_k<<<ceil_div(ROWS * 128, 256), 256, 0, st>>>(t1, t2, CAT2, ROWS * 128);
  launch_gemm(st, CAT2, W1, b1, nullptr, O1, ROWS, 128, 128, 128, 128, 128, 1, 0, 0, 0, 1);
  launch_gemm(st, O1, W2, b2, nullptr, O2, ROWS, 64, 128, 128, 64, 64, 1, 0, 0, 0, 0);
  chstat_k<<<64, 256, 0, st>>>(O2, STAT, ROWS);
  bn_apply_k<<<ceil_div(ROWS * 64, 256), 256, 0, st>>>(O2, STAT, bg, bb, out, ROWS * 64);
}

// --------------------------------- entry -----------------------------------

extern "C" void kernel_launch(void* const* d_in, const int* in_sizes, int n_in,
                              void* d_out, int out_size, void* d_ws, size_t ws_size,
                              hipStream_t stream) {
  (void)in_sizes; (void)n_in; (void)out_size; (void)ws_size;
  float* ws = (float*)d_ws;
  const float* x   = (const float*)d_in[0];
  const float* STE = (const float*)d_in[1];
  const float* adj = (const float*)d_in[2];
  float* out = (float*)d_out;

  // 1) x_enc = x + STE ; sp = transpose(x_enc) to (B,T,N,C)
  add_k<<<ceil_div(ROWS * 64, 256), 256, 0, stream>>>(x, STE, ws + O_XENC, ROWS * 64);
  tr_to_sp_k<<<ceil_div(ROWS * 64, 256), 256, 0, stream>>>(ws + O_XENC, ws + O_SP, ROWS * 64);

  // 2) bi-Mamba -> XM   (mamba operates on raw x)
  run_mamba(stream, x, d_in, 3, 0, ws, ws + A_YF);   // fwd
  run_mamba(stream, x, d_in, 13, 1, ws, ws + A_YB);  // bwd (time reversed)
  mamba_cat_k<<<ceil_div(ROWS * 128, 256), 256, 0, stream>>>(x, ws + A_YF, ws + A_YB,
                                                             ws + A_CAT, ROWS * 128, Tc);
  launch_gemm(stream, ws + A_CAT, (const float*)d_in[23], (const float*)d_in[24], nullptr,
              ws + A_YPROJ, ROWS, 64, 128, 128, 64, 64, 1, 0, 0, 0, 1);
  ln_k<<<ROWS, 64, 0, stream>>>(x, ws + A_YPROJ, (const float*)d_in[25],
                                (const float*)d_in[26], ws + O_XM);

  // 3) temporal attention -> XT (rows layout, batch=SEQ sequences of length T)
  run_tf(stream, ws + O_XENC, SEQ, Tc, d_in, 27, nullptr, ws, ws + O_XT);

  // 4) GCN on sp -> XG (sp layout)
  ahat_k<<<Nc, 256, 0, stream>>>(adj, ws + O_AHAT, Nc);
  {
    const float* cur = ws + O_SP;
    float *T0 = ws + A_T0, *M0 = ws + A_M0, *G0 = ws + A_G0;
    float* dst[2] = { ws + A_H1, ws + O_XG };
    for (int it = 0; it < 2; it++) {
      const float* Wi  = (const float*)d_in[51 + it * 4 + 0];
      const float* bi  = (const float*)d_in[51 + it * 4 + 1];
      const float* Wgi = (const float*)d_in[51 + it * 4 + 2];
      const float* bgi = (const float*)d_in[51 + it * 4 + 3];
      launch_gemm(stream, cur, Wi, nullptr, nullptr, T0, ROWS, 64, 64, 64, 64, 64,
                  1, 0, 0, 0, 0);
      // m = relu(A_hat @ t + b), batched over B*T slices
      launch_gemm(stream, ws + O_AHAT, T0, bi, nullptr, M0, Nc, 64, Nc, Nc, 64, 64,
                  Bc * Tc, 0, (long long)Nc * 64, (long long)Nc * 64, 1);
      launch_gemm(stream, cur, Wgi, bgi, nullptr, G0, ROWS, 64, 64, 64, 64, 64,
                  1, 0, 0, 0, 4);
      gcomb_k<<<ceil_div(ROWS * 64, 256), 256, 0, stream>>>(G0, M0, cur, dst[it], ROWS * 64);
      cur = dst[it];
    }
  }

  // 5) spatial attention (masked) on sp -> XSAT (sp layout)
  run_tf(stream, ws + O_SP, Bc * Tc, Nc, d_in, 39, adj, ws, ws + O_XSAT);

  // 6) TFN spatial: x_s = BN(relu(cat(xg, xsat) @ W1 + b1) @ W2 + b2)
  run_tfn(stream, ws + O_XG, ws + O_XSAT, 1, d_in, 59, ws, ws + O_XS);

  // 7) TFN temporal: x_t from (xm, xt)
  run_tfn(stream, ws + O_XM, ws + O_XT, 0, d_in, 65, ws, ws + O_XTF);

  // 8) out = x + relu(cat(x_s, x_t) @ gate_W + gate_b)
  cat_rows_k<<<ceil_div(ROWS * 128, 256), 256, 0, stream>>>(ws + O_XS, ws + O_XTF,
                                                            ws + A_CATG, ROWS * 128);
  launch_gemm(stream, ws + A_CATG, (const float*)d_in[71], (const float*)d_in[72],
              x, out, ROWS, 64, 128, 128, 64, 64, 1, 0, 0, 0, 1);
}